// VQVAE_64476049047766
// MI455X (gfx1250) — compile-verified
//
#include <hip/hip_runtime.h>

#define DECAY 0.9f
#define GAIN  (1.0f - DECAY)

typedef __attribute__((ext_vector_type(2))) float v2f;
typedef __attribute__((ext_vector_type(4))) float v4f;
typedef __attribute__((ext_vector_type(8))) float v8f;

// Problem constants (from reference setup_inputs):
//   batch=2048, cw_dim=512, dim_codes=128, dict_size=512, dim_embedding=4

// ---------------------------------------------------------------------------
// Kernel 0: zero the hist/accum scratch (327,680 floats, contiguous)
// ---------------------------------------------------------------------------
__global__ __launch_bounds__(256) void vq_init_kernel(float* __restrict__ p, int n) {
    int i = blockIdx.x * 256 + threadIdx.x;
    if (i < n) p[i] = 0.0f;
}

// ---------------------------------------------------------------------------
// Kernel 1: nearest-code assignment via V_WMMA_F32_16X16X4_F32.
// Score = x·d - 0.5*||d||^2  (argmax over s == argmin of L2 distance).
// grid = (128 codes, 8 chunks of 256 batch rows), block = 256 threads (8 waves).
// Final per-row argmax uses an LDS transpose (ds_store_b64/ds_load_b64 +
// branchless selects) instead of shuffle butterflies -> no exec-mask ladders.
// ---------------------------------------------------------------------------
__global__ __launch_bounds__(256) void vq_assign_kernel(
    const float* __restrict__ x,        // (2048, 512) == (2048, 128, 4)
    const float* __restrict__ dict,     // (128, 512, 4)
    int*   __restrict__ idx_ws,         // (2048, 128)
    float* __restrict__ hist,           // (128, 512)
    float* __restrict__ accum,          // (128, 512, 4)
    float* __restrict__ cw_embed)       // (2048, 512)
{
    __shared__ __align__(16) float dict_lds[512 * 4];          // dictionary[c]
    __shared__ float bias_lds[512];                            // -0.5*||d_s||^2
    __shared__ int   idx_lds[256];                             // per-row winner
    __shared__ __align__(8) unsigned long long pair_lds[8 * 16 * 16]; // (val,idx)

    const int c         = blockIdx.x;        // code dim 0..127
    const int chunkBase = blockIdx.y * 256;  // batch chunk
    const int t    = threadIdx.x;
    const int lane = t & 31;
    const int wave = t >> 5;
    const int half = lane >> 4;              // 0: K/M group 0, 1: group 1
    const int l15  = lane & 15;

    // ---- stage dictionary[c] into LDS (2048 floats, 8 per thread) ----
    {
        const v4f* src = (const v4f*)(dict + (size_t)c * 512 * 4);
        v4f* dst = (v4f*)dict_lds;
        dst[t]       = src[t];
        dst[t + 256] = src[t + 256];
    }
    __syncthreads();
    for (int s = t; s < 512; s += 256) {
        float d0 = dict_lds[s * 4 + 0], d1 = dict_lds[s * 4 + 1];
        float d2 = dict_lds[s * 4 + 2], d3 = dict_lds[s * 4 + 3];
        bias_lds[s] = -0.5f * (d0 * d0 + d1 * d1 + d2 * d2 + d3 * d3);
    }
    __syncthreads();

    // ---- each wave handles 2 batch tiles of 16 rows ----
    for (int tt = 0; tt < 2; ++tt) {
        const int tileBase = chunkBase + (wave * 2 + tt) * 16;

        // A (16x4 f32): lanes 0-15 hold M=lane K={0,1}; lanes 16-31 K={2,3}
        v2f a;
        {
            const float* xp = x + (size_t)(tileBase + l15) * 512 + c * 4 + half * 2;
            a.x = xp[0];
            a.y = xp[1];
        }

        float best_val[8];
        int   best_idx[8];
        #pragma unroll
        for (int v = 0; v < 8; ++v) { best_val[v] = -3.4e38f; best_idx[v] = 0; }

        for (int s0 = 0; s0 < 512; s0 += 16) {
            const int sc = s0 + l15;   // this lane's column (code index)
            // B (4x16 f32): lane column N=l15; lanes 0-15 K={0,1}, 16-31 K={2,3}
            v2f b;
            b.x = dict_lds[sc * 4 + half * 2 + 0];
            b.y = dict_lds[sc * 4 + half * 2 + 1];
            const float cb = bias_lds[sc];
            v8f cacc;
            #pragma unroll
            for (int v = 0; v < 8; ++v) cacc[v] = cb;

            // D = A x B + C  -> v_wmma_f32_16x16x4_f32
            v8f d = __builtin_amdgcn_wmma_f32_16x16x4_f32(
                false, a, false, b, (short)0, cacc, false, false);

            #pragma unroll
            for (int v = 0; v < 8; ++v) {
                if (d[v] > best_val[v]) { best_val[v] = d[v]; best_idx[v] = sc; }
            }
        }

        // ---- LDS transpose: lane holds rows {v+8*half}, column l15 ----
        #pragma unroll
        for (int v = 0; v < 8; ++v) {
            const int m = v + 8 * half;
            unsigned long long pk =
                (unsigned long long)__float_as_uint(best_val[v]) |
                ((unsigned long long)(unsigned)best_idx[v] << 32);
            pair_lds[(wave * 16 + m) * 16 + l15] = pk;
        }
        __syncthreads();   // uniform: all 256 threads execute the tt loop

        if (lane < 16) {   // lane m scans the 16 candidates of row m
            float bv = -3.4e38f;
            int   bi = 0;
            #pragma unroll
            for (int n = 0; n < 16; ++n) {
                unsigned long long pk = pair_lds[(wave * 16 + l15) * 16 + n];
                float v = __uint_as_float((unsigned)(pk & 0xffffffffu));
                int   i = (int)(pk >> 32);
                // branchless: strictly-greater OR (equal AND smaller index)
                int take = (int)(v > bv) | ((int)(v == bv) & (int)(i < bi));
                bv = take ? v : bv;
                bi = take ? i : bi;
            }
            const int row = tileBase + l15;
            idx_ws[(size_t)row * 128 + c] = bi;
            idx_lds[row - chunkBase]      = bi;
        }
        __syncthreads();   // pair_lds reused by next tile
    }

    // ---- pass 2: gather cw_embed + scatter hist/accum (1 row per thread) ----
    {
        const int row = chunkBase + t;
        const int s   = idx_lds[t];
        const float* xp = x + (size_t)row * 512 + c * 4;
        const float x0 = xp[0], x1 = xp[1], x2 = xp[2], x3 = xp[3];

        atomicAdd(&hist[(size_t)c * 512 + s], 1.0f);
        float* ap = accum + ((size_t)c * 512 + s) * 4;
        atomicAdd(ap + 0, x0); atomicAdd(ap + 1, x1);
        atomicAdd(ap + 2, x2); atomicAdd(ap + 3, x3);

        v4f e = *(const v4f*)(dict_lds + s * 4);
        *(v4f*)(cw_embed + (size_t)row * 512 + c * 4) = e;
    }
}

// ---------------------------------------------------------------------------
// Kernel 2: EMA update of counts and dictionary (one thread per (c,s))
// ---------------------------------------------------------------------------
__global__ __launch_bounds__(256) void vq_finalize_kernel(
    const float* __restrict__ dict, const float* __restrict__ ema,
    const float* __restrict__ hist, const float* __restrict__ accum,
    float* __restrict__ new_dict, float* __restrict__ new_counts)
{
    const int i = blockIdx.x * 256 + threadIdx.x;   // (c*512 + s), 65536 total
    const float nc = DECAY * ema[i] + GAIN * hist[i];
    new_counts[i] = nc;
    const float g = GAIN / nc;
    v4f dv = *(const v4f*)(dict  + (size_t)i * 4);
    v4f av = *(const v4f*)(accum + (size_t)i * 4);
    v4f nd = dv * DECAY + av * g;
    *(v4f*)(new_dict + (size_t)i * 4) = nd;
}

// ---------------------------------------------------------------------------
// Kernel 3: one-hot streaming writer. One wave per (b,c) row of 512 floats.
// 512 MB total -> bandwidth-bound; b128 non-temporal stores.
// ---------------------------------------------------------------------------
__global__ __launch_bounds__(256) void vq_onehot_kernel(
    const int* __restrict__ idx_ws, float* __restrict__ onehot)
{
    const int wave = threadIdx.x >> 5;
    const int lane = threadIdx.x & 31;
    const long long row = (long long)blockIdx.x * 8 + wave;  // b*128 + c
    const int s_sel = idx_ws[row];
    v4f* out = (v4f*)(onehot + row * 512);
    #pragma unroll
    for (int i = 0; i < 4; ++i) {
        const int sb = (i * 32 + lane) * 4;
        v4f v;
        v.x = (sb + 0 == s_sel) ? 1.0f : 0.0f;
        v.y = (sb + 1 == s_sel) ? 1.0f : 0.0f;
        v.z = (sb + 2 == s_sel) ? 1.0f : 0.0f;
        v.w = (sb + 3 == s_sel) ? 1.0f : 0.0f;
        __builtin_nontemporal_store(v, &out[i * 32 + lane]);
    }
}

extern "C" void kernel_launch(void* const* d_in, const int* in_sizes, int n_in,
                              void* d_out, int out_size, void* d_ws, size_t ws_size,
                              hipStream_t stream) {
    const float* x    = (const float*)d_in[0];   // (2048, 512)
    const float* dict = (const float*)d_in[1];   // (128, 512, 4)
    const float* ema  = (const float*)d_in[2];   // (128, 512)

    // Outputs concatenated flat in return order.
    float* out        = (float*)d_out;
    float* cw_embed   = out;                                  // 2048*512
    float* onehot     = out + 1048576;                        // 2048*128*512
    float* new_dict   = out + 1048576 + 134217728;            // 128*512*4
    float* new_counts = new_dict + 262144;                    // 128*512

    // Workspace: idx (1 MB) | hist (256 KB) | accum (1 MB)
    char*  ws     = (char*)d_ws;
    int*   idx_ws = (int*)ws;
    float* hist   = (float*)(ws + 1048576);
    float* accum  = hist + 65536;

    // 1) zero hist+accum (327,680 floats = 1280 blocks * 256)
    vq_init_kernel<<<1280, 256, 0, stream>>>(hist, 327680);

    // 2) WMMA nearest-code assignment + cw_embed + scatter accumulation
    dim3 agrid(128, 8);
    vq_assign_kernel<<<agrid, 256, 0, stream>>>(x, dict, idx_ws, hist, accum, cw_embed);

    // 3) EMA finalize (65,536 entries)
    vq_finalize_kernel<<<256, 256, 0, stream>>>(dict, ema, hist, accum, new_dict, new_counts);

    // 4) one-hot streaming write (262,144 rows, 8 rows/block)
    vq_onehot_kernel<<<32768, 256, 0, stream>>>(idx_ws, onehot);
}